// NeuralVMEmbedding_83391085019705
// MI455X (gfx1250) — compile-verified
//
#include <hip/hip_runtime.h>
#include <hip/hip_bf16.h>
#include <stdint.h>

// ---- problem constants (from reference) ----
#define BATCH      16
#define SEQ        8192
#define DM         512
#define VOCAB      272
#define CODE_START 256
#define CODE_END   257
#define MEM_TOK    258
#define ADDR_KEY   206
#define MEM_STORE  455

typedef float v4f __attribute__((ext_vector_type(4)));

// =====================================================================
// Pass 2 (hot kernel, placed first for disasm visibility):
// gather + mask + stream out.  One 512-float row per wave per iteration;
// embedding row gathered global->LDS with CDNA5 async loads (ASYNCcnt),
// wave-uniform 1.0 overrides poked directly into the LDS copy, then the
// row streamed to global with non-temporal b128 stores.
// =====================================================================
__device__ __forceinline__ void issue_row_async(uint64_t embed_base, int tok,
                                                unsigned ldsBufByte, int lane) {
  // Make sure all our previous ds reads of this buffer have completed
  // before the async engine may rewrite it.
  asm volatile("s_wait_dscnt 0" ::: "memory");
  const unsigned g0 = (unsigned)tok * (DM * 4u) + (unsigned)lane * 16u;
  const unsigned l0 = ldsBufByte + (unsigned)lane * 16u;
#pragma unroll
  for (int k = 0; k < 4; ++k) {
    // GVS form: per-lane 32-bit byte offset + 64-bit SGPR base.
    asm volatile("global_load_async_to_lds_b128 %0, %1, %2"
                 :: "v"(l0 + 512u * (unsigned)k),
                    "v"(g0 + 512u * (unsigned)k),
                    "s"(embed_base)
                 : "memory");
  }
}

__device__ __forceinline__ void load_ctx(int r, const int* __restrict__ tok_ids,
                                         const int* __restrict__ cs_arr,
                                         const int* __restrict__ end_arr, int memEnd,
                                         int& tok, int& clo, int& chi, int& ctop, int& cmem) {
  const int b = r >> 13;           // r / SEQ
  const int s = r & (SEQ - 1);     // r % SEQ
  tok = __builtin_amdgcn_readfirstlane(tok_ids[r]);
  const int cs = __builtin_amdgcn_readfirstlane(cs_arr[r]);
  const int e  = __builtin_amdgcn_readfirstlane(end_arr[b]);
  const int addr  = s - cs - 1;
  const bool valid = (cs >= 0) && (s < e) && (tok < 256) && (addr >= 0);
  clo  = valid ? (ADDR_KEY +       (addr & 15))        : -1;
  chi  = valid ? (ADDR_KEY + 16 + ((addr >> 4) & 15))  : -1;
  ctop = valid ? (ADDR_KEY + 32 + ((addr >> 8) & 15))  : -1;
  cmem = (tok == MEM_TOK && s < memEnd) ? MEM_STORE : -1;
}

__global__ __launch_bounds__(256) void vm_embed_kernel(
    const int* __restrict__ tok_ids, const float* __restrict__ embed,
    const int* __restrict__ cs_arr, const int* __restrict__ end_arr,
    const int* __restrict__ memend_p, float* __restrict__ out) {
  __shared__ float lds[8 * 2 * DM];                       // 8 waves x double buffer = 32 KB
  const int lane = (int)(threadIdx.x & 31u);
  const int wave = (int)(threadIdx.x >> 5);
  const int nwaves = (int)((gridDim.x * blockDim.x) >> 5);
  int r = (int)((blockIdx.x * blockDim.x + threadIdx.x) >> 5);

  float* const wbuf = &lds[(size_t)wave * 2 * DM];
  // Low 32 bits of the generic LDS pointer == workgroup-relative LDS byte addr.
  const unsigned ldsBase = (unsigned)(uintptr_t)&lds[0] + (unsigned)wave * (2u * DM * 4u);
  const uint64_t embed_u = (uint64_t)(uintptr_t)embed;
  const int memEnd = __builtin_amdgcn_readfirstlane(*memend_p);
  const int TOTAL = BATCH * SEQ;
  if (r >= TOTAL) return;

  int tok0, clo0, chi0, ctop0, cmem0;
  load_ctx(r, tok_ids, cs_arr, end_arr, memEnd, tok0, clo0, chi0, ctop0, cmem0);
  issue_row_async(embed_u, tok0, ldsBase, lane);          // prolog: buffer 0
  int p = 0;

  while (true) {
    const int rn = r + nwaves;
    const bool hasNext = rn < TOTAL;                      // wave-uniform
    int tok1 = 0, clo1 = -1, chi1 = -1, ctop1 = -1, cmem1 = -1;
    if (hasNext) {
      load_ctx(rn, tok_ids, cs_arr, end_arr, memEnd, tok1, clo1, chi1, ctop1, cmem1);
      issue_row_async(embed_u, tok1, ldsBase + (unsigned)((p ^ 1) * (DM * 4)), lane);
    }
    // 4 async ops per row, loads complete in order:
    // <=4 outstanding means the current row has landed in LDS.
    if (hasNext) asm volatile("s_wait_asynccnt 4" ::: "memory");
    else         asm volatile("s_wait_asynccnt 0" ::: "memory");

    // Wave-uniform 1.0 overrides: 4 lanes poke the LDS copy (VALU-free masking).
    float* buf = wbuf + p * DM;
    const int ch = (lane == 0) ? clo0 : (lane == 1) ? chi0 : (lane == 2) ? ctop0 : cmem0;
    if (lane < 4 && ch >= 0) buf[ch] = 1.0f;

    // Stream LDS -> global, non-temporal (256 MB output, never re-read).
    const v4f* src = (const v4f*)(buf + lane * 16);
    float* drow = out + (size_t)r * DM + (size_t)(lane * 16);
    const v4f a0 = src[0], a1 = src[1], a2 = src[2], a3 = src[3];
    __builtin_nontemporal_store(a0, (v4f*)(drow + 0));
    __builtin_nontemporal_store(a1, (v4f*)(drow + 4));
    __builtin_nontemporal_store(a2, (v4f*)(drow + 8));
    __builtin_nontemporal_store(a3, (v4f*)(drow + 12));

    if (!hasNext) break;
    r = rn; tok0 = tok1; clo0 = clo1; chi0 = chi1; ctop0 = ctop1; cmem0 = cmem1;
    p ^= 1;
  }
}

// =====================================================================
// Pass 1: per batch-row prefix scan.
//   cs[b,s]  = inclusive cummax over j<=s of (tok[b,j]==CODE_START ? j : -1)
//   end[b]   = min over j of (tok[b,j]==CODE_END ? j : SEQ)
// One block (1024 threads) per batch row; 8 chunks of 1024 with a carried
// max.  Rolled log-step loop (unrolling it only bloats I-cache; the scan
// is barrier-bound and <1% of total work).
// =====================================================================
__global__ __launch_bounds__(1024) void vm_scan_kernel(const int* __restrict__ tok,
                                                       int* __restrict__ cs_out,
                                                       int* __restrict__ end_out) {
  __shared__ int sdata[1024];
  __shared__ int s_end;
  __shared__ int s_carry;
  const int b = blockIdx.x;
  const int t = (int)threadIdx.x;
  if (t == 0) { s_end = SEQ; s_carry = -1; }
  __syncthreads();
  const int* row = tok + (size_t)b * SEQ;
  for (int c = 0; c < SEQ; c += 1024) {
    const int s  = c + t;
    const int tk = row[s];
    int v = (tk == CODE_START) ? s : -1;
    sdata[t] = v;
    __syncthreads();
    for (int off = 1; off < 1024; off <<= 1) {
      const int u = (t >= off) ? sdata[t - off] : -1;
      __syncthreads();
      v = (v > u) ? v : u;
      sdata[t] = v;
      __syncthreads();
    }
    const int cs = (v > s_carry) ? v : s_carry;   // reads carry from previous chunk
    cs_out[(size_t)b * SEQ + s] = cs;
    if (tk == CODE_END) atomicMin(&s_end, s);
    __syncthreads();                              // all reads of s_carry done
    if (t == 1023) s_carry = cs;
    __syncthreads();
  }
  if (t == 0) end_out[b] = s_end;
}

// =====================================================================
// Host launcher.
// Inputs: d_in[0]=token_ids (int[16*8192]), d_in[1]=embed_weight (float[272*512]),
//         d_in[2]=mem_history_end (int[1]).  Output: float[16*8192*512].
// Workspace: cs[16*8192] ints + end[16] ints = 524,352 bytes.
// =====================================================================
extern "C" void kernel_launch(void* const* d_in, const int* in_sizes, int n_in,
                              void* d_out, int out_size, void* d_ws, size_t ws_size,
                              hipStream_t stream) {
  (void)in_sizes; (void)n_in; (void)out_size; (void)ws_size;
  const int*   tok    = (const int*)d_in[0];
  const float* embed  = (const float*)d_in[1];
  const int*   memend = (const int*)d_in[2];
  float*       out    = (float*)d_out;
  int*         cs     = (int*)d_ws;
  int*         endi   = cs + BATCH * SEQ;

  vm_scan_kernel<<<BATCH, 1024, 0, stream>>>(tok, cs, endi);
  // 2048 blocks x 8 waves = 16384 waves; 131072 rows => 8 rows/wave.
  vm_embed_kernel<<<2048, 256, 0, stream>>>(tok, embed, cs, endi, memend, out);
}